// gNNC_8589934592546
// MI455X (gfx1250) — compile-verified
//
#include <hip/hip_runtime.h>
#include <math.h>

typedef float v2f __attribute__((ext_vector_type(2)));
typedef float v8f __attribute__((ext_vector_type(8)));

#define TPB 256

__device__ __forceinline__ v8f wmma_f32(v2f a, v2f b, v8f c) {
  // D = A(16x4) * B(4x16) + C, fp32, wave32
  return __builtin_amdgcn_wmma_f32_16x16x4_f32(false, a, false, b, (short)0, c,
                                               false, false);
}

__device__ __forceinline__ float block_reduce_sum(float v, float* red) {
  int tid = threadIdx.x;
  red[tid] = v;
  __syncthreads();
  for (int s = TPB / 2; s > 0; s >>= 1) {
    if (tid < s) red[tid] += red[tid + s];
    __syncthreads();
  }
  float r = red[0];
  __syncthreads();
  return r;
}

// ---------------------------------------------------------------------------
// Generic batched fp32 WMMA GEMM:
//   C[b, m, n] = epi( alpha * sum_k A[b?, m, k] * B[b, k, n] ) * mask[b, n]
//                (+ C_old if accumulate)
// Block = 256 threads = 8 waves covering a 128(M) x 128(N) macro-tile.
// Each wave: 32(M) x 64(N) = 2x4 fragment block, 8 accumulators.
// Inner loop: 12 loads per 8 WMMAs (1.5 loads/WMMA).
// ---------------------------------------------------------------------------
__global__ __launch_bounds__(TPB) void gemm_f32_wmma(
    const float* __restrict__ A, long aBatchStride, int aRowStride, int aColStride,
    const float* __restrict__ Bm, long bBatchStride, int ldb,
    float* __restrict__ C, long cBatchStride, int ldc,
    int K,
    const float* __restrict__ alphaPtr, int alphaIdx,
    const float* __restrict__ maskPtr, int maskStride,
    int accumulate, int epiTanh) {
  const int b    = blockIdx.z;
  const int n0   = blockIdx.x * 128;
  const int m0   = blockIdx.y * 128;
  const int tid  = threadIdx.x;
  const int w    = tid >> 5;
  const int lane = tid & 31;
  const int lh   = lane >> 4;   // lane-half select
  const int lm   = lane & 15;
  const int mi   = m0 + (w >> 1) * 32;   // wave's two M tiles: mi, mi+16
  const int nj   = n0 + (w & 1) * 64;    // wave's four N tiles: nj + {0,16,32,48}

  const float* Ab = A + (long)b * aBatchStride;
  const float* Bb = Bm + (long)b * bBatchStride;
  float*       Cb = C + (long)b * cBatchStride;

  v8f acc[2][4] = {};
  const float* aRow0 = Ab + (long)(mi + lm) * aRowStride;
  const float* aRow1 = Ab + (long)(mi + 16 + lm) * aRowStride;
  for (int k = 0; k < K; k += 4) {
    const int ka = k + 2 * lh;
    v2f a0, a1;
    a0.x = aRow0[(long)ka * aColStride];
    a0.y = aRow0[(long)(ka + 1) * aColStride];
    a1.x = aRow1[(long)ka * aColStride];
    a1.y = aRow1[(long)(ka + 1) * aColStride];
    const float* bR0 = Bb + (long)ka * ldb + lm;
    const float* bR1 = bR0 + ldb;
    v2f bf[4];
#pragma unroll
    for (int j = 0; j < 4; ++j) {
      bf[j].x = bR0[nj + j * 16];
      bf[j].y = bR1[nj + j * 16];
    }
#pragma unroll
    for (int j = 0; j < 4; ++j) {
      acc[0][j] = wmma_f32(a0, bf[j], acc[0][j]);
      acc[1][j] = wmma_f32(a1, bf[j], acc[1][j]);
    }
  }

  const float alpha = alphaPtr ? alphaPtr[alphaIdx] : 1.0f;
#pragma unroll
  for (int j = 0; j < 4; ++j) {
    const int nn = nj + j * 16 + lm;
    const float mk = maskPtr ? maskPtr[(long)b * maskStride + nn] : 1.0f;
#pragma unroll
    for (int mt = 0; mt < 2; ++mt) {
#pragma unroll
      for (int v = 0; v < 8; ++v) {
        const int mm = mi + mt * 16 + v + 8 * lh;
        const long idx = (long)mm * ldc + nn;
        float val = acc[mt][j][v] * alpha;
        if (epiTanh) val = tanhf(val);
        val *= mk;
        if (accumulate) val += Cb[idx];
        Cb[idx] = val;
      }
    }
  }
}

// ---------------------------------------------------------------------------
// Gram + distance: Wd[b,n,m] = sqrt(max(3*(n2[n]+n2[m]-2*Zc^T Zc)/kchan, eps))
//                              * mask[n]*mask[m];  also accumulate sum(Wd).
// Same 128x128 / wave-2x4 blocking as the GEMM.
// ---------------------------------------------------------------------------
__global__ __launch_bounds__(TPB) void gram_dist_kernel(
    const float* __restrict__ Zc,   // B x KP x N
    const float* __restrict__ n2,   // B x N
    const float* __restrict__ mask, // B x N
    float* __restrict__ Wd,         // B x N x N
    float* __restrict__ sig,        // B (atomic accumulator, pre-zeroed)
    int N, int KP, float kchan) {
  const int b    = blockIdx.z;
  const int m0   = blockIdx.x * 128;
  const int n0   = blockIdx.y * 128;
  const int tid  = threadIdx.x;
  const int w    = tid >> 5;
  const int lane = tid & 31;
  const int lh   = lane >> 4;
  const int lm   = lane & 15;
  const int ni   = n0 + (w >> 1) * 32;  // rows: ni, ni+16
  const int mj   = m0 + (w & 1) * 64;   // cols: mj + {0,16,32,48}

  const float* Z = Zc + (long)b * KP * N;
  v8f acc[2][4] = {};
  for (int k = 0; k < KP; k += 4) {
    const float* r0 = Z + (long)(k + 2 * lh) * N;
    v2f a0, a1;
    a0.x = r0[ni + lm];
    a0.y = r0[N + ni + lm];
    a1.x = r0[ni + 16 + lm];
    a1.y = r0[N + ni + 16 + lm];
    v2f bf[4];
#pragma unroll
    for (int j = 0; j < 4; ++j) {
      bf[j].x = r0[mj + j * 16 + lm];
      bf[j].y = r0[N + mj + j * 16 + lm];
    }
#pragma unroll
    for (int j = 0; j < 4; ++j) {
      acc[0][j] = wmma_f32(a0, bf[j], acc[0][j]);
      acc[1][j] = wmma_f32(a1, bf[j], acc[1][j]);
    }
  }

  const float* n2b = n2 + (long)b * N;
  const float* mk  = mask + (long)b * N;
  float*       Wb  = Wd + (long)b * N * N;
  float local = 0.f;
#pragma unroll
  for (int j = 0; j < 4; ++j) {
    const int m = mj + j * 16 + lm;
    const float n2m = n2b[m];
    const float mkm = mk[m];
#pragma unroll
    for (int mt = 0; mt < 2; ++mt) {
#pragma unroll
      for (int v = 0; v < 8; ++v) {
        const int n = ni + mt * 16 + v + 8 * lh;
        float d  = 3.0f * (n2b[n] + n2m - 2.0f * acc[mt][j][v]) / kchan;
        float wd = sqrtf(fmaxf(d, 1e-12f)) * mk[n] * mkm;
        Wb[(long)n * N + m] = wd;
        local += wd;
      }
    }
  }
  __shared__ float red[TPB];
  float s = block_reduce_sum(local, red);
  if (tid == 0) atomicAdd(&sig[b], s);
}

// ---------------------------------------------------------------------------
// Build Zc = centered( [normalize_masked(Z) ; pos] ), padded to KP rows.
// One block per (channel, batch); row fits in LDS.
// ---------------------------------------------------------------------------
__global__ __launch_bounds__(TPB) void build_zc_kernel(
    const float* __restrict__ Z, const float* __restrict__ mask,
    const float* __restrict__ cnt, float* __restrict__ Zc,
    int C, int N, int KP) {
  const int c = blockIdx.x;
  const int b = blockIdx.y;
  const int tid = threadIdx.x;
  float* out = Zc + ((long)b * KP + c) * N;
  if (c > C) {  // zero padding rows
    for (int i = tid; i < N; i += TPB) out[i] = 0.f;
    return;
  }
  __shared__ float row[1024];
  __shared__ float red[TPB];
  const float* mk = mask + (long)b * N;
  const float ct = cnt[b];
  if (c < C) {
    const float* zr = Z + ((long)b * C + c) * N;
    for (int i = tid; i < N; i += TPB) row[i] = zr[i];
    __syncthreads();
    float s1 = 0.f;
    for (int i = tid; i < N; i += TPB) s1 += row[i] * mk[i];
    s1 = block_reduce_sum(s1, red);
    const float muv = s1 / ct;
    for (int i = tid; i < N; i += TPB) row[i] = row[i] - muv * mk[i];
    __syncthreads();
    float s2 = 0.f;
    for (int i = tid; i < N; i += TPB) s2 += row[i] * row[i];
    s2 = block_reduce_sum(s2, red);
    const float dn = sqrtf(s2 / ct + 1e-4f);
    for (int i = tid; i < N; i += TPB) row[i] = row[i] / dn;
    __syncthreads();
  } else {  // c == C : position channel
    for (int i = tid; i < N; i += TPB)
      row[i] = 0.5f * ((float)i / (float)(N - 1)) * mk[i];
    __syncthreads();
  }
  // plain-mean centering (tr2dist_small)
  float s3 = 0.f;
  for (int i = tid; i < N; i += TPB) s3 += row[i];
  s3 = block_reduce_sum(s3, red);
  const float m2 = s3 / (float)N;
  for (int i = tid; i < N; i += TPB) out[i] = row[i] - m2;
}

__global__ void maskcnt_kernel(const float* __restrict__ mask,
                               float* __restrict__ cnt, int N) {
  __shared__ float red[TPB];
  const int b = blockIdx.x;
  float s = 0.f;
  for (int i = threadIdx.x; i < N; i += TPB) s += mask[(long)b * N + i];
  s = block_reduce_sum(s, red);
  if (threadIdx.x == 0) cnt[b] = s;
}

__global__ void n2_kernel(const float* __restrict__ Zc, float* __restrict__ n2,
                          int N, int KP) {
  const long idx = (long)blockIdx.x * TPB + threadIdx.x;  // b*N + n
  const int b = (int)(idx / N);
  const int n = (int)(idx % N);
  const float* Zb = Zc + (long)b * KP * N + n;
  float s = 0.f;
  for (int k = 0; k < KP; ++k) {
    const float v = Zb[(long)k * N];
    s += v * v;
  }
  n2[idx] = s;
}

// Wd_raw -> L = -exp(-Wd/sigma*10)*mm (in place); Wsum[b,m] = column sum.
__global__ void exp_lap_kernel(float* __restrict__ G,
                               const float* __restrict__ mask,
                               const float* __restrict__ sig,
                               const float* __restrict__ cnt,
                               float* __restrict__ Wsum, int N) {
  const int b = blockIdx.y;
  const int m = blockIdx.x * TPB + threadIdx.x;
  const float ct = cnt[b];
  const float sigma = sig[b] / (ct * ct);
  const float sc = 10.0f / sigma;
  const float mkm = mask[(long)b * N + m];
  const float* mk = mask + (long)b * N;
  float* Gb = G + (long)b * N * N;
  float s = 0.f;
  for (int n = 0; n < N; ++n) {
    const long id = (long)n * N + m;
    const float wf = expf(-Gb[id] * sc) * mk[n] * mkm;
    Gb[id] = -wf;
    s += wf;
  }
  Wsum[(long)b * N + m] = s;
}

__global__ void diag_kernel(float* __restrict__ G,
                            const float* __restrict__ Wsum, int N) {
  const long idx = (long)blockIdx.x * TPB + threadIdx.x;  // b*N + n
  const int b = (int)(idx / N);
  const int n = (int)(idx % N);
  G[(long)b * N * N + (long)n * N + n] += Wsum[idx];
}

__global__ void masked_mean_kernel(const float* __restrict__ A,
                                   const float* __restrict__ mask,
                                   const float* __restrict__ cnt,
                                   float* __restrict__ mu, int C, int N) {
  const int c = blockIdx.x;
  const int b = blockIdx.y;
  __shared__ float red[TPB];
  const float* row = A + ((long)b * C + c) * N;
  const float* mk = mask + (long)b * N;
  float s = 0.f;
  for (int i = threadIdx.x; i < N; i += TPB) s += row[i] * mk[i];
  s = block_reduce_sum(s, red);
  if (threadIdx.x == 0) mu[(long)b * C + c] = s / cnt[b];
}

// Ai = relu( ((Ai - mu)*m) / sqrt(sum_ch(((Ai-mu)*m)^2) + 0.001) ), in place.
__global__ void normrelu_kernel(float* __restrict__ A,
                                const float* __restrict__ mask,
                                const float* __restrict__ mu, int C, int N) {
  const int b = blockIdx.y;
  const int n = blockIdx.x * TPB + threadIdx.x;
  const float mkn = mask[(long)b * N + n];
  float* Ab = A + (long)b * C * N;
  const float* mub = mu + (long)b * C;
  float s = 0.f;
  for (int o = 0; o < C; ++o) {
    const float x = (Ab[(long)o * N + n] - mub[o]) * mkn;
    s += x * x;
  }
  const float d = sqrtf(s + 0.001f);
  for (int o = 0; o < C; ++o) {
    const float x = (Ab[(long)o * N + n] - mub[o]) * mkn / d;
    Ab[(long)o * N + n] = fmaxf(x, 0.f);
  }
}

__global__ void axpy_kernel(float* __restrict__ Y, const float* __restrict__ X,
                            float a, long n) {
  const long i = (long)blockIdx.x * TPB + threadIdx.x;
  if (i < n) Y[i] += a * X[i];
}

// Zout[b,o,n] = (sum_c Wend[o,c]*Z[b,c,n]) * mask[b,n]   (o in 0..2)
__global__ void zout_kernel(const float* __restrict__ Zcur,
                            const float* __restrict__ Wend,
                            const float* __restrict__ mask,
                            float* __restrict__ out, int C, int N) {
  const long idx = (long)blockIdx.x * TPB + threadIdx.x;  // b*3*N + o*N + n
  const int b = (int)(idx / (3 * N));
  const long r = idx % (3 * N);
  const int o = (int)(r / N);
  const int n = (int)(r % N);
  const float* Zb = Zcur + (long)b * C * N;
  const float* w = Wend + (long)o * C;
  float s = 0.f;
  for (int c = 0; c < C; ++c) s += w[c] * Zb[(long)c * N + n];
  out[idx] = s * mask[(long)b * N + n];
}

__global__ void zmean3_kernel(const float* __restrict__ Zout,
                              float* __restrict__ mean3, int N) {
  const int o = blockIdx.x;
  const int b = blockIdx.y;
  __shared__ float red[TPB];
  const float* row = Zout + ((long)b * 3 + o) * N;
  float s = 0.f;
  for (int i = threadIdx.x; i < N; i += TPB) s += row[i];
  s = block_reduce_sum(s, red);
  if (threadIdx.x == 0) mean3[b * 3 + o] = s / (float)N;
}

__global__ void zcenter_kernel(const float* __restrict__ Zout,
                               const float* __restrict__ mean3,
                               float* __restrict__ Zoc,
                               float* __restrict__ n2o, int N) {
  const long idx = (long)blockIdx.x * TPB + threadIdx.x;  // b*N + n
  const int b = (int)(idx / N);
  const int n = (int)(idx % N);
  float s = 0.f;
  for (int o = 0; o < 3; ++o) {
    const long id = ((long)b * 3 + o) * N + n;
    const float v = Zout[id] - mean3[b * 3 + o];
    Zoc[id] = v;
    s += v * v;
  }
  n2o[idx] = s;
}

__global__ void distout_kernel(const float* __restrict__ Zoc,
                               const float* __restrict__ n2o,
                               float* __restrict__ dout, int N) {
  const long idx = (long)blockIdx.x * TPB + threadIdx.x;  // b*N*N + n*N + m
  const long NN = (long)N * N;
  const int b = (int)(idx / NN);
  const long r = idx % NN;
  const int n = (int)(r / N);
  const int m = (int)(r % N);
  const float* Zb = Zoc + (long)b * 3 * N;
  const float dot =
      Zb[n] * Zb[m] + Zb[N + n] * Zb[N + m] + Zb[2 * N + n] * Zb[2 * N + m];
  const float* nb = n2o + (long)b * N;
  const float d = nb[n] + nb[m] - 2.0f * dot;  // k=3 -> 3*d/3 = d
  dout[idx] = sqrtf(fmaxf(d, 1e-12f));
}

// ---------------------------------------------------------------------------
extern "C" void kernel_launch(void* const* d_in, const int* in_sizes, int n_in,
                              void* d_out, int out_size, void* d_ws,
                              size_t ws_size, hipStream_t stream) {
  (void)in_sizes; (void)n_in; (void)out_size; (void)ws_size;
  const int B = 16, CIN = 40, C = 256, N = 1024, KP = 272, NL = 6;
  const float h = 0.01f;

  const float* Zin  = (const float*)d_in[0];  // B x 40 x N
  const float* mask = (const float*)d_in[1];  // B x N
  const float* K0   = (const float*)d_in[2];  // 256 x 40
  const float* Knet = (const float*)d_in[3];  // 6 x 3
  const float* W    = (const float*)d_in[4];  // 6 x 3 x 256 x 256
  const float* Wend = (const float*)d_in[5];  // 3 x 256

  float* ws = (float*)d_ws;
  const long BCN = (long)B * C * N;
  float* Zcur  = ws;
  float* bufA  = Zcur + BCN;
  float* bufT1 = bufA + BCN;
  float* bufT2 = bufT1 + BCN;
  float* Zc    = bufT2 + BCN;                 // B*KP*N
  float* Gbuf  = Zc + (long)B * KP * N;       // B*N*N
  float* sm    = Gbuf + (long)B * N * N;
  float* cnt   = sm;                // B
  float* sig   = cnt + B;           // B
  float* n2    = sig + B;           // B*N
  float* Wsum  = n2 + (long)B * N;  // B*N
  float* mu    = Wsum + (long)B * N;        // B*C
  float* mean3 = mu + (long)B * C;          // B*3
  float* n2o   = mean3 + (long)B * 3;       // B*N
  float* Zoc   = n2o + (long)B * N;         // B*3*N

  float* outDist = (float*)d_out;                       // B*N*N
  float* outZ    = outDist + (long)B * N * N;           // B*3*N

  const dim3 gGemm(N / 128, C / 128, B);  // 8 x 2 x B
  const dim3 gGram(N / 128, N / 128, B);  // 8 x 8 x B

  hipLaunchKernelGGL(maskcnt_kernel, dim3(B), dim3(TPB), 0, stream, mask, cnt, N);

  // Zcur = tanh(K0 @ Zin) * m   (M=256, K=40, N=1024)
  hipLaunchKernelGGL(gemm_f32_wmma, gGemm, dim3(TPB), 0, stream,
                     K0, 0L, CIN, 1,
                     Zin, (long)CIN * N, N,
                     Zcur, BCN / B, N,
                     CIN, (const float*)nullptr, 0, mask, N, 0, 1);

  for (int i = 0; i < NL; ++i) {
    const float* Wi = W + (long)i * 3 * C * C;

    // ---- graph Laplacian ----
    hipLaunchKernelGGL(build_zc_kernel, dim3(KP, B), dim3(TPB), 0, stream,
                       Zcur, mask, cnt, Zc, C, N, KP);
    hipLaunchKernelGGL(n2_kernel, dim3((B * N) / TPB), dim3(TPB), 0, stream,
                       Zc, n2, N, KP);
    hipMemsetAsync(sig, 0, B * sizeof(float), stream);
    hipLaunchKernelGGL(gram_dist_kernel, gGram, dim3(TPB), 0, stream,
                       Zc, n2, mask, Gbuf, sig, N, KP, 257.0f);
    hipLaunchKernelGGL(exp_lap_kernel, dim3(N / TPB, B), dim3(TPB), 0, stream,
                       Gbuf, mask, sig, cnt, Wsum, N);
    hipLaunchKernelGGL(diag_kernel, dim3((B * N) / TPB), dim3(TPB), 0, stream,
                       Gbuf, Wsum, N);

    // ---- first half: bufA = K0*cv(W0,Z) + (K1*cv(W1,Z))@L + ((K2*cv(W2,Z))@L)@L
    hipLaunchKernelGGL(gemm_f32_wmma, gGemm, dim3(TPB), 0, stream,
                       Wi + 0L * C * C, 0L, C, 1, Zcur, BCN / B, N,
                       bufA, BCN / B, N, C, Knet, i * 3 + 0, mask, N, 0, 0);
    hipLaunchKernelGGL(gemm_f32_wmma, gGemm, dim3(TPB), 0, stream,
                       Wi + 1L * C * C, 0L, C, 1, Zcur, BCN / B, N,
                       bufT1, BCN / B, N, C, Knet, i * 3 + 1, mask, N, 0, 0);
    hipLaunchKernelGGL(gemm_f32_wmma, gGemm, dim3(TPB), 0, stream,
                       bufT1, BCN / B, N, 1, Gbuf, (long)N * N, N,
                       bufA, BCN / B, N, N, (const float*)nullptr, 0,
                       (const float*)nullptr, 0, 1, 0);
    hipLaunchKernelGGL(gemm_f32_wmma, gGemm, dim3(TPB), 0, stream,
                       Wi + 2L * C * C, 0L, C, 1, Zcur, BCN / B, N,
                       bufT1, BCN / B, N, C, Knet, i * 3 + 2, mask, N, 0, 0);
    hipLaunchKernelGGL(gemm_f32_wmma, gGemm, dim3(TPB), 0, stream,
                       bufT1, BCN / B, N, 1, Gbuf, (long)N * N, N,
                       bufT2, BCN / B, N, N, (const float*)nullptr, 0,
                       (const float*)nullptr, 0, 0, 0);
    hipLaunchKernelGGL(gemm_f32_wmma, gGemm, dim3(TPB), 0, stream,
                       bufT2, BCN / B, N, 1, Gbuf, (long)N * N, N,
                       bufA, BCN / B, N, N, (const float*)nullptr, 0,
                       (const float*)nullptr, 0, 1, 0);

    // ---- normalize + relu (in place on bufA) ----
    hipLaunchKernelGGL(masked_mean_kernel, dim3(C, B), dim3(TPB), 0, stream,
                       bufA, mask, cnt, mu, C, N);
    hipLaunchKernelGGL(normrelu_kernel, dim3(N / TPB, B), dim3(TPB), 0, stream,
                       bufA, mask, mu, C, N);

    // ---- second half with transposed weights (A = bufA) ----
    hipLaunchKernelGGL(gemm_f32_wmma, gGemm, dim3(TPB), 0, stream,
                       Wi + 0L * C * C, 0L, 1, C, bufA, BCN / B, N,
                       bufT1, BCN / B, N, C, Knet, i * 3 + 0, mask, N, 0, 0);
    hipLaunchKernelGGL(gemm_f32_wmma, gGemm, dim3(TPB), 0, stream,
                       Wi + 1L * C * C, 0L, 1, C, bufA, BCN / B, N,
                       bufT2, BCN / B, N, C, Knet, i * 3 + 1, mask, N, 0, 0);
    hipLaunchKernelGGL(gemm_f32_wmma, gGemm, dim3(TPB), 0, stream,
                       bufT2, BCN / B, N, 1, Gbuf, (long)N * N, N,
                       bufT1, BCN / B, N, N, (const float*)nullptr, 0,
                       (const float*)nullptr, 0, 1, 0);
    hipLaunchKernelGGL(gemm_f32_wmma, gGemm, dim3(TPB), 0, stream,
                       Wi + 2L * C * C, 0L, 1, C, bufA, BCN / B, N,
                       bufT2, BCN / B, N, C, Knet, i * 3 + 2, mask, N, 0, 0);
    hipLaunchKernelGGL(gemm_f32_wmma, gGemm, dim3(TPB), 0, stream,
                       bufT2, BCN / B, N, 1, Gbuf, (long)N * N, N,
                       bufA, BCN / B, N, N, (const float*)nullptr, 0,
                       (const float*)nullptr, 0, 0, 0);
    hipLaunchKernelGGL(gemm_f32_wmma, gGemm, dim3(TPB), 0, stream,
                       bufA, BCN / B, N, 1, Gbuf, (long)N * N, N,
                       bufT1, BCN / B, N, N, (const float*)nullptr, 0,
                       (const float*)nullptr, 0, 1, 0);

    // ---- Z = Z - h * Ai ----
    hipLaunchKernelGGL(axpy_kernel, dim3((unsigned)(BCN / TPB)), dim3(TPB), 0,
                       stream, Zcur, bufT1, -h, BCN);
  }

  // ---- output head ----
  hipLaunchKernelGGL(zout_kernel, dim3((B * 3 * N) / TPB), dim3(TPB), 0, stream,
                     Zcur, Wend, mask, outZ, C, N);
  hipLaunchKernelGGL(zmean3_kernel, dim3(3, B), dim3(TPB), 0, stream,
                     outZ, mean3, N);
  hipLaunchKernelGGL(zcenter_kernel, dim3((B * N) / TPB), dim3(TPB), 0, stream,
                     outZ, mean3, Zoc, n2o, N);
  hipLaunchKernelGGL(distout_kernel,
                     dim3((unsigned)(((long)B * N * N) / TPB)), dim3(TPB), 0,
                     stream, Zoc, n2o, outDist, N);
}